// L1_ChamferLoss_82746839925382
// MI455X (gfx1250) — compile-verified
//
#include <hip/hip_runtime.h>
#include <hip/hip_bf16.h>
#include <math.h>

typedef __attribute__((ext_vector_type(2))) float v2f;
typedef __attribute__((ext_vector_type(8))) float v8f;

// ---------------------------------------------------------------------------
// Prep: build augmented point arrays for the GEMM formulation.
//   rowAug[i] = { x, y, z, 1 }           (A-side, 16x4 fragment source)
//   colAug[i] = { -2x, -2y, -2z, |p|^2 } (B-side, 4x16 fragment source)
//   sq[i]     = |p|^2                    (added back after the row-min)
// ---------------------------------------------------------------------------
__global__ void chamfer_prep_kernel(const float* __restrict__ in,
                                    float* __restrict__ rowAug,
                                    float* __restrict__ colAug,
                                    float* __restrict__ sq,
                                    int total)
{
    int i = blockIdx.x * blockDim.x + threadIdx.x;
    if (i >= total) return;
    float x = in[3 * (size_t)i + 0];
    float y = in[3 * (size_t)i + 1];
    float z = in[3 * (size_t)i + 2];
    float s = fmaf(x, x, fmaf(y, y, z * z));
    float4 r = make_float4(x, y, z, 1.0f);
    float4 c = make_float4(-2.0f * x, -2.0f * y, -2.0f * z, s);
    *(float4*)(rowAug + 4 * (size_t)i) = r;
    *(float4*)(colAug + 4 * (size_t)i) = c;
    sq[i] = s;
}

__global__ void chamfer_zero_kernel(float* __restrict__ out)
{
    if (threadIdx.x == 0 && blockIdx.x == 0) out[0] = 0.0f;
}

// ---------------------------------------------------------------------------
// One Chamfer direction. Each wave owns a 16-row tile of the "row" point set
// and streams every 16-column tile of the "col" set through
// v_wmma_f32_16x16x4_f32:  G = A' x B'  with  A'[r]=(x,y,z,1),
// B'[c]=(-2x,-2y,-2z,|b|^2)  =>  G[r,c] = |b|^2 - 2 a.b.
// dist[r] = max(|a|^2 + min_c G[r,c], 0); accumulate sum(sqrt(dist))*scale.
//
// Inner loop is software-pipelined: the b64 loads for column-tile pair t+1
// are issued before the WMMAs consuming pair t, so each load gets a full
// iteration of latency cover; one running lane pointer (+512B/iter) lets the
// two loads share an address register with 0/256-byte immediate offsets.
// ---------------------------------------------------------------------------
__global__ void chamfer_dir_kernel(const float* __restrict__ rowA,   // [B][NR][4] {x,y,z,1}
                                   const float* __restrict__ colB,   // [B][NC][4] {-2x,-2y,-2z,s}
                                   const float* __restrict__ sqA,    // [B][NR]
                                   int NR, int NC, int numWaves,
                                   float scale,
                                   float* __restrict__ out)
{
    const int lane = threadIdx.x & 31;
    const int wave = blockIdx.x * (blockDim.x >> 5) + (threadIdx.x >> 5);
    if (wave >= numWaves) return;                     // wave-uniform: EXEC stays all-1s

    const int tilesPerBatch = NR >> 4;
    const int batch   = wave / tilesPerBatch;
    const int rowBase = (wave % tilesPerBatch) << 4;

    const float* rp = rowA + (size_t)batch * NR * 4;
    const float* cp = colB + (size_t)batch * NC * 4;
    const float* sp = sqA  + (size_t)batch * NR;

    const int half = lane >> 4;    // 0: K={0,1}, 1: K={2,3}
    const int l15  = lane & 15;

    // A fragment (16x4 f32): 2 VGPRs per lane, reused for all column tiles.
    v2f a = *(const v2f*)(rp + ((size_t)(rowBase + l15) << 2) + (half << 1));

    const float INF = 3.402823466e38f;
    v8f mins = { INF, INF, INF, INF, INF, INF, INF, INF };

    // Per-lane running pointer into the column-side augmented array.
    const float* bp = cp + ((size_t)l15 << 2) + (half << 1);

    const int pairs = NC >> 5;        // number of 32-column double-tiles
    int cDone = pairs << 5;

    if (pairs > 0) {
        // Prologue: load the first double-tile.
        v2f b0 = *(const v2f*)(bp);
        v2f b1 = *(const v2f*)(bp + 64);   // +16 points = +256 bytes
        bp += 128;                         // +32 points = +512 bytes

        for (int t = 1; t < pairs; ++t) {
            // Issue next double-tile's loads before this tile's WMMAs.
            v2f n0 = *(const v2f*)(bp);
            v2f n1 = *(const v2f*)(bp + 64);
            __builtin_prefetch(bp + 256, 0, 3);
            bp += 128;

            v8f g0 = {};
            v8f g1 = {};
            g0 = __builtin_amdgcn_wmma_f32_16x16x4_f32(false, a, false, b0,
                                                       (short)0, g0, false, false);
            g1 = __builtin_amdgcn_wmma_f32_16x16x4_f32(false, a, false, b1,
                                                       (short)0, g1, false, false);
#pragma unroll
            for (int i = 0; i < 8; ++i) {
                mins[i] = fminf(mins[i], fminf(g0[i], g1[i]));
            }
            b0 = n0;
            b1 = n1;
        }

        // Epilogue: drain the last double-tile.
        v8f g0 = {};
        v8f g1 = {};
        g0 = __builtin_amdgcn_wmma_f32_16x16x4_f32(false, a, false, b0,
                                                   (short)0, g0, false, false);
        g1 = __builtin_amdgcn_wmma_f32_16x16x4_f32(false, a, false, b1,
                                                   (short)0, g1, false, false);
#pragma unroll
        for (int i = 0; i < 8; ++i) {
            mins[i] = fminf(mins[i], fminf(g0[i], g1[i]));
        }
    }

    // Generic remainder (NC % 32), one 16-column tile at a time.
    for (int c0 = cDone; c0 < NC; c0 += 16) {
        v2f b = *(const v2f*)(cp + ((size_t)(c0 + l15) << 2) + (half << 1));
        v8f g = {};
        g = __builtin_amdgcn_wmma_f32_16x16x4_f32(false, a, false, b,
                                                  (short)0, g, false, false);
#pragma unroll
        for (int i = 0; i < 8; ++i) mins[i] = fminf(mins[i], g[i]);
    }

    // C/D layout: VGPR i holds row rowBase+i (lanes 0-15) and rowBase+i+8
    // (lanes 16-31), col = lane&15. XOR-reduce the min within each 16-lane half.
    float localSum = 0.0f;
#pragma unroll
    for (int i = 0; i < 8; ++i) {
        float m = mins[i];
        m = fminf(m, __shfl_xor(m, 1, 32));
        m = fminf(m, __shfl_xor(m, 2, 32));
        m = fminf(m, __shfl_xor(m, 4, 32));
        m = fminf(m, __shfl_xor(m, 8, 32));
        if (l15 == 0) {
            int row = rowBase + i + (half << 3);
            float d = fmaxf(sp[row] + m, 0.0f);
            localSum += sqrtf(d);
        }
    }
    if (l15 == 0) {
        atomicAdd(out, localSum * scale);
    }
}

// ---------------------------------------------------------------------------
extern "C" void kernel_launch(void* const* d_in, const int* in_sizes, int n_in,
                              void* d_out, int out_size, void* d_ws, size_t ws_size,
                              hipStream_t stream)
{
    const float* a1 = (const float*)d_in[0];   // [B, N, 3] f32
    const float* a2 = (const float*)d_in[1];   // [B, M, 3] f32

    const int B = 4;                           // per reference setup
    const int N = in_sizes[0] / (B * 3);
    const int M = in_sizes[1] / (B * 3);

    float* ws = (float*)d_ws;
    size_t o = 0;
    float* rowAug1 = ws + o; o += (size_t)B * N * 4;
    float* colAug1 = ws + o; o += (size_t)B * N * 4;
    float* sq1     = ws + o; o += (size_t)B * N;
    float* rowAug2 = ws + o; o += (size_t)B * M * 4;
    float* colAug2 = ws + o; o += (size_t)B * M * 4;
    float* sq2     = ws + o; o += (size_t)B * M;

    float* out = (float*)d_out;

    const int t1 = B * N;
    const int t2 = B * M;
    chamfer_prep_kernel<<<(t1 + 255) / 256, 256, 0, stream>>>(a1, rowAug1, colAug1, sq1, t1);
    chamfer_prep_kernel<<<(t2 + 255) / 256, 256, 0, stream>>>(a2, rowAug2, colAug2, sq2, t2);
    chamfer_zero_kernel<<<1, 32, 0, stream>>>(out);

    // Direction 1: rows = array1, cols = array2  -> mean(sqrt(dist1)) / 2
    {
        int numWaves = (B * N) / 16;
        int blocks   = (numWaves * 32 + 255) / 256;
        float scale  = 1.0f / (2.0f * (float)B * (float)N);
        chamfer_dir_kernel<<<blocks, 256, 0, stream>>>(rowAug1, colAug2, sq1,
                                                       N, M, numWaves, scale, out);
    }
    // Direction 2: rows = array2, cols = array1  -> mean(sqrt(dist2)) / 2
    {
        int numWaves = (B * M) / 16;
        int blocks   = (numWaves * 32 + 255) / 256;
        float scale  = 1.0f / (2.0f * (float)B * (float)M);
        chamfer_dir_kernel<<<blocks, 256, 0, stream>>>(rowAug2, colAug1, sq2,
                                                       M, N, numWaves, scale, out);
    }
}